// L1_SSIM_Loss_52639119180250
// MI455X (gfx1250) — compile-verified
//
#include <hip/hip_runtime.h>

typedef float v2f __attribute__((ext_vector_type(2)));
typedef float v8f __attribute__((ext_vector_type(8)));

#define NVOX 8192000   // 2*160*160*160
#define SY   160
#define SZ   25600
#define SN   4096000
#define BLOCKS 4000    // 32000 tiles / 8 waves per 256-thread block

// 11-tap Gaussian (sigma=1.5, normalized), padded with zeros so that
// GWP[d + 15] is valid for every d = i - n that can occur (d in [-15, 27]).
// Band w[0..10] lives at indices 15..25.
__constant__ float GWP[44] = {
    0.0f, 0.0f, 0.0f, 0.0f, 0.0f, 0.0f, 0.0f, 0.0f, 0.0f, 0.0f,
    0.0f, 0.0f, 0.0f, 0.0f, 0.0f,
    0.00102838f, 0.00759875f, 0.03600077f, 0.10936080f, 0.21300560f,
    0.26601180f, 0.21300560f, 0.10936080f, 0.03600077f, 0.00759875f,
    0.00102838f,
    0.0f, 0.0f, 0.0f, 0.0f, 0.0f, 0.0f, 0.0f, 0.0f, 0.0f, 0.0f,
    0.0f, 0.0f, 0.0f, 0.0f, 0.0f, 0.0f, 0.0f, 0.0f};

// Per-lane banded-weight operands for all 7 K-chunks, computed once.
// Unconditional loads (table is padded), reused across every WMMA.
static __device__ __forceinline__ void make_weights(v2f wv[7], int mn, int khalf) {
#pragma unroll
  for (int c = 0; c < 7; ++c) {
    const int d = c * 4 + khalf - mn + 15;
    wv[c].x = GWP[d];
    wv[c].y = GWP[d + 1];
  }
}

// Tile conv via V_WMMA_F32_16X16X4_F32:
//  DATA_IN_A=1:  D(16x16) = Data(16x26) * W(26x16)      (N = conv axis)
//  DATA_IN_A=0:  D(16x16) = W^T(16x26) * Data(26x16)    (M = conv axis, N = free)
// ld(p): data at conv-axis position p for this lane's row/col (zero-padded,
// implemented branch-free by the caller).
template <bool DATA_IN_A, typename L>
static __device__ __forceinline__ v8f conv_tile(L&& ld, const v2f wv[7], int j0,
                                                int khalf) {
  v8f acc = {};
#pragma unroll
  for (int c = 0; c < 7; ++c) {
    const int i0 = c * 4 + khalf;
    v2f d;
    d.x = ld(j0 - 5 + i0);
    d.y = ld(j0 - 5 + i0 + 1);
    const v2f a = DATA_IN_A ? d : wv[c];
    const v2f b = DATA_IN_A ? wv[c] : d;
    acc = __builtin_amdgcn_wmma_f32_16x16x4_f32(false, a, false, b, (short)0,
                                                acc, false, false);
  }
  return acc;
}

// Branch-free zero-padded sample: clamp index, load unconditionally, select 0.
static __device__ __forceinline__ int clamp160(int p) {
  return (p < 0) ? 0 : ((p > 159) ? 159 : p);
}

// Pass 1: blur along x (stride 1). Field value generated on the fly from p,t.
template <int FIELD>
__global__ void __launch_bounds__(256) k_blur_x(const float* __restrict__ P,
                                                const float* __restrict__ T,
                                                float* __restrict__ dst) {
  const int wave = blockIdx.x * 8 + (threadIdx.x >> 5);
  const int lane = threadIdx.x & 31;
  const int mn = lane & 15;
  const int khalf = (lane >> 4) * 2;
  int r = wave;
  const int nb = r / 16000; r -= nb * 16000;
  const int z  = r / 100;   r -= z * 100;
  const int yt = r / 10;
  const int xt = r - yt * 10;
  const int base = nb * SN + z * SZ + yt * 16 * SY;
  const int j0 = xt * 16;

  v2f wv[7];
  make_weights(wv, mn, khalf);

  const int rowbase = base + mn * SY;
  auto ld = [&](int p) -> float {
    const int idx = rowbase + clamp160(p);
    const float a = P[idx];
    const float b = T[idx];
    float v;
    if (FIELD == 0)      v = a;
    else if (FIELD == 1) v = b;
    else if (FIELD == 2) v = a * a;
    else if (FIELD == 3) v = b * b;
    else                 v = a * b;
    return ((unsigned)p < 160u) ? v : 0.0f;
  };
  v8f d = conv_tile<true>(ld, wv, j0, khalf);

  const int mb = (lane >> 4) * 8;
#pragma unroll
  for (int rr = 0; rr < 8; ++rr)
    dst[base + (rr + mb) * SY + j0 + mn] = d[rr];   // contiguous along x
}

// Pass 2: blur along y (stride 160). Free axis = x (coalesced loads & stores).
__global__ void __launch_bounds__(256) k_blur_y(const float* __restrict__ src,
                                                float* __restrict__ dst) {
  const int wave = blockIdx.x * 8 + (threadIdx.x >> 5);
  const int lane = threadIdx.x & 31;
  const int mn = lane & 15;
  const int khalf = (lane >> 4) * 2;
  int r = wave;
  const int nb = r / 16000; r -= nb * 16000;
  const int z  = r / 100;   r -= z * 100;
  const int xt = r / 10;
  const int yt = r - xt * 10;
  const int base = nb * SN + z * SZ + xt * 16;
  const int j0 = yt * 16;

  v2f wv[7];
  make_weights(wv, mn, khalf);

  auto ld = [&](int p) -> float {
    const float v = src[base + mn + clamp160(p) * SY];
    return ((unsigned)p < 160u) ? v : 0.0f;
  };
  v8f d = conv_tile<false>(ld, wv, j0, khalf);

  const int mb = (lane >> 4) * 8;
#pragma unroll
  for (int rr = 0; rr < 8; ++rr)
    dst[base + mn + (j0 + rr + mb) * SY] = d[rr];
}

// Pass 3 (fused): blur along z for all 5 fields, SSIM map, L1, block reduction.
__global__ void __launch_bounds__(256) k_blur_z_ssim(const float* __restrict__ F,
                                                     const float* __restrict__ P,
                                                     const float* __restrict__ T,
                                                     float* __restrict__ parts,
                                                     int nblocks) {
  const int wave = blockIdx.x * 8 + (threadIdx.x >> 5);
  const int lane = threadIdx.x & 31;
  const int mn = lane & 15;
  const int khalf = (lane >> 4) * 2;
  int r = wave;
  const int nb = r / 16000; r -= nb * 16000;
  const int y  = r / 100;   r -= y * 100;
  const int xt = r / 10;
  const int zt = r - xt * 10;
  const int base = nb * SN + y * SY + xt * 16;
  const int j0 = zt * 16;

  v2f wv[7];
  make_weights(wv, mn, khalf);

  v8f acc[5] = {};
#pragma unroll
  for (int c = 0; c < 7; ++c) {
    const int p0 = j0 - 5 + c * 4 + khalf;
    const int p1 = p0 + 1;
    const int o0 = base + mn + clamp160(p0) * SZ;
    const int o1 = base + mn + clamp160(p1) * SZ;
    const bool v0 = (unsigned)p0 < 160u;
    const bool v1 = (unsigned)p1 < 160u;
#pragma unroll
    for (int f = 0; f < 5; ++f) {
      const int fo = f * NVOX;
      v2f b;
      b.x = v0 ? F[fo + o0] : 0.0f;   // load only when in range, else 0
      b.y = v1 ? F[fo + o1] : 0.0f;
      acc[f] = __builtin_amdgcn_wmma_f32_16x16x4_f32(false, wv[c], false, b,
                                                     (short)0, acc[f], false,
                                                     false);
    }
  }

  const float C1v = 1.0e-4f;   // (0.01*1)^2
  const float C2v = 9.0e-4f;   // (0.03*1)^2
  float ssum = 0.0f, lsum = 0.0f;
  const int mb = (lane >> 4) * 8;
#pragma unroll
  for (int rr = 0; rr < 8; ++rr) {
    const int zz = j0 + rr + mb;
    const int idx = base + mn + zz * SZ;
    const float mu1 = acc[0][rr], mu2 = acc[1][rr];
    const float bpp = acc[2][rr], btt = acc[3][rr], bpt = acc[4][rr];
    const float mu1s = mu1 * mu1, mu2s = mu2 * mu2, m12 = mu1 * mu2;
    const float s1 = bpp - mu1s, s2 = btt - mu2s, s12 = bpt - m12;
    const float num = (2.0f * m12 + C1v) * (2.0f * s12 + C2v);
    const float den = (mu1s + mu2s + C1v) * (s1 + s2 + C2v) + 1.0e-12f;
    ssum += num / den;
    lsum += fabsf(P[idx] - T[idx]);
  }

  __shared__ float sh1[256];
  __shared__ float sh2[256];
  const int tid = threadIdx.x;
  sh1[tid] = ssum; sh2[tid] = lsum;
  __syncthreads();
  for (int off = 128; off > 0; off >>= 1) {
    if (tid < off) { sh1[tid] += sh1[tid + off]; sh2[tid] += sh2[tid + off]; }
    __syncthreads();
  }
  if (tid == 0) {
    parts[blockIdx.x] = sh1[0];
    parts[nblocks + blockIdx.x] = sh2[0];
  }
}

// Deterministic final fold (fixed order, single block).
__global__ void __launch_bounds__(256) k_finalize(const float* __restrict__ parts,
                                                  int nparts,
                                                  float* __restrict__ out) {
  __shared__ float sh1[256];
  __shared__ float sh2[256];
  const int tid = threadIdx.x;
  float a = 0.0f, b = 0.0f;
  for (int i = tid; i < nparts; i += 256) {
    a += parts[i];
    b += parts[nparts + i];
  }
  sh1[tid] = a; sh2[tid] = b;
  __syncthreads();
  for (int off = 128; off > 0; off >>= 1) {
    if (tid < off) { sh1[tid] += sh1[tid + off]; sh2[tid] += sh2[tid + off]; }
    __syncthreads();
  }
  if (tid == 0) {
    const float inv = 1.0f / 8192000.0f;
    const float ssim = sh1[0] * inv;
    const float l1 = sh2[0] * inv;
    out[0] = 0.7f * l1 + 0.3f * (1.0f - ssim);
  }
}

extern "C" void kernel_launch(void* const* d_in, const int* in_sizes, int n_in,
                              void* d_out, int out_size, void* d_ws, size_t ws_size,
                              hipStream_t stream) {
  (void)in_sizes; (void)n_in; (void)out_size; (void)ws_size;
  const float* P = (const float*)d_in[0];
  const float* T = (const float*)d_in[1];
  float* ws = (float*)d_ws;
  float* bufXY = ws;                        // 5*NVOX: xy-blurred fields
  float* tmp   = ws + (size_t)5 * NVOX;     // NVOX: x-blurred scratch
  float* parts = ws + (size_t)6 * NVOX;     // 2*BLOCKS partial sums

  // Per-field x then y blur (ping-pong through tmp).
  k_blur_x<0><<<BLOCKS, 256, 0, stream>>>(P, T, tmp);
  k_blur_y<<<BLOCKS, 256, 0, stream>>>(tmp, bufXY + (size_t)0 * NVOX);
  k_blur_x<1><<<BLOCKS, 256, 0, stream>>>(P, T, tmp);
  k_blur_y<<<BLOCKS, 256, 0, stream>>>(tmp, bufXY + (size_t)1 * NVOX);
  k_blur_x<2><<<BLOCKS, 256, 0, stream>>>(P, T, tmp);
  k_blur_y<<<BLOCKS, 256, 0, stream>>>(tmp, bufXY + (size_t)2 * NVOX);
  k_blur_x<3><<<BLOCKS, 256, 0, stream>>>(P, T, tmp);
  k_blur_y<<<BLOCKS, 256, 0, stream>>>(tmp, bufXY + (size_t)3 * NVOX);
  k_blur_x<4><<<BLOCKS, 256, 0, stream>>>(P, T, tmp);
  k_blur_y<<<BLOCKS, 256, 0, stream>>>(tmp, bufXY + (size_t)4 * NVOX);

  // Fused z blur + SSIM + L1 + block reduction, then deterministic fold.
  k_blur_z_ssim<<<BLOCKS, 256, 0, stream>>>(bufXY, P, T, parts, BLOCKS);
  k_finalize<<<1, 256, 0, stream>>>(parts, BLOCKS, (float*)d_out);
}